// AINN_SC_Merged_10007273799763
// MI455X (gfx1250) — compile-verified
//
#include <hip/hip_runtime.h>
#include <hip/hip_bf16.h>
#include <math.h>

typedef __attribute__((ext_vector_type(2))) float v2f;
typedef __attribute__((ext_vector_type(8))) float v8f;

#define B_    8
#define T_    61
#define F_    13
#define KCLS_ 8
#define R_    128          // K * B_ref = 8*16
#define N_    136          // B_ + R_
#define NT_   (N_*T_)      // 8296 rows
#define H_    96
#define D_    32
#define P_    (B_*R_)      // 1024 DP problems
#define PADV  8.0f

__device__ __forceinline__ float gelu_exact(float x) {
  return 0.5f * x * (1.0f + erff(x * 0.70710678118654752f));
}

// ---------------------------------------------------------------- projection
// h0[row, c] = sum_f src[row, f] * prj_w[c, f] + prj_b[c]
__global__ void k_proj(const float* __restrict__ x, const float* __restrict__ refs,
                       const float* __restrict__ w, const float* __restrict__ b,
                       float* __restrict__ out) {
  int row = blockIdx.x;          // 0..NT_-1
  int c   = threadIdx.x;         // 0..95
  int n = row / T_, t = row % T_;
  const float* src = (n < B_) ? (x + (n * T_ + t) * F_)
                              : (refs + ((n - B_) * T_ + t) * F_);
  float acc = b[c];
#pragma unroll
  for (int f = 0; f < F_; ++f) acc += src[f] * w[c * F_ + f];
  out[row * H_ + c] = acc;
}

// ------------------------------------------------------------ depthwise conv
// zdw[n,t,c] = sum_{k=0..6} h[n, t+(k-3)*dil, c] * dw[c,k] + db[c]  (zero pad)
__global__ void k_dwconv(const float* __restrict__ h, const float* __restrict__ dw,
                         const float* __restrict__ db, float* __restrict__ out,
                         int dil) {
  int idx = blockIdx.x * blockDim.x + threadIdx.x;
  if (idx >= NT_ * H_) return;
  int c = idx % H_;
  int row = idx / H_;
  int t = row % T_, n = row / T_;
  float acc = db[c];
#pragma unroll
  for (int k = 0; k < 7; ++k) {
    int tt = t + (k - 3) * dil;
    float v = (tt >= 0 && tt < T_) ? h[(n * T_ + tt) * H_ + c] : 0.0f;
    acc += v * dw[c * 7 + k];
  }
  out[idx] = acc;
}

// ------------------------------------------- pointwise conv GEMM via f32 WMMA
// out[m, o] = gelu( sum_c A[m,c] * W[o,c] + bias[o] )  (+ res[m,o])
// One wave = one 16x16 tile; all 24 K-step fragments preloaded into registers
// (branch-free: row index clamped; out-of-range D rows discarded at store),
// then a back-to-back V_WMMA_F32_16X16X4_F32 accumulation chain.
template <int NN, bool ADDRES>
__global__ void k_pw_wmma(const float* __restrict__ A,     // [NT_, 96]
                          const float* __restrict__ W,     // [NN, 96]
                          const float* __restrict__ bias,  // [NN]
                          const float* __restrict__ res,   // [NT_, NN] or null
                          float* __restrict__ out) {       // [NT_, NN]
  const int MT = (NT_ + 15) / 16;   // 519
  const int NTL = NN / 16;
  int wid = (blockIdx.x * blockDim.x + threadIdx.x) >> 5;
  if (wid >= MT * NTL) return;
  int tm = wid / NTL, tn = wid % NTL;
  int lane = threadIdx.x & 31;
  int hlf = lane >> 4;              // selects K sub-pair (A) / K row pair (B)
  int l16 = lane & 15;
  int m  = tm * 16 + l16;           // A row for this lane
  if (m > NT_ - 1) m = NT_ - 1;     // clamp: keeps EXEC uniform, no branches
  int nn = tn * 16 + l16;           // B (weight) row for this lane
  float bcol = bias[tn * 16 + l16]; // C/D column = lane&15
  v8f acc;
#pragma unroll
  for (int v = 0; v < 8; ++v) acc[v] = bcol;

  v2f aReg[24], bReg[24];
#pragma unroll
  for (int k4 = 0; k4 < 24; ++k4) {
    int ka = k4 * 4 + hlf * 2;                       // even -> 8B aligned
    aReg[k4] = *(const v2f*)(A + (size_t)m * 96 + ka);
    bReg[k4] = *(const v2f*)(W + (size_t)nn * 96 + ka);
  }
#pragma unroll
  for (int k4 = 0; k4 < 24; ++k4)
    acc = __builtin_amdgcn_wmma_f32_16x16x4_f32(false, aReg[k4], false, bReg[k4],
                                                (short)0, acc, false, false);
#pragma unroll
  for (int v = 0; v < 8; ++v) {
    int rrow = tm * 16 + hlf * 8 + v;   // D-matrix row for vgpr v
    if (rrow < NT_) {
      float val = gelu_exact(acc[v]);
      if constexpr (ADDRES) val += res[rrow * NN + tn * 16 + l16];
      out[rrow * NN + tn * 16 + l16] = val;
    }
  }
}

// --------------------------------------------------------------- layer norm
__global__ void k_ln(const float* __restrict__ in, const float* __restrict__ w,
                     const float* __restrict__ b, float* __restrict__ out) {
  int wid = (blockIdx.x * blockDim.x + threadIdx.x) >> 5;
  if (wid >= NT_) return;
  int lane = threadIdx.x & 31;
  float v = in[wid * D_ + lane];
  float s = v;
#pragma unroll
  for (int o = 16; o > 0; o >>= 1) s += __shfl_xor(s, o, 32);
  float mu = s * (1.0f / 32.0f);
  float d = v - mu;
  float sq = d * d;
#pragma unroll
  for (int o = 16; o > 0; o >>= 1) sq += __shfl_xor(sq, o, 32);
  float var = sq * (1.0f / 32.0f);
  out[wid * D_ + lane] = d * rsqrtf(var + 1e-5f) * w[lane] + b[lane];
}

// --------------------------- L2-normalize rows; apply symmetric W to queries
__global__ void k_normqw(const float* __restrict__ enc, const float* __restrict__ W,
                         float* __restrict__ qW, float* __restrict__ kkn) {
  int wid = (blockIdx.x * blockDim.x + threadIdx.x) >> 5;
  if (wid >= NT_) return;
  int lane = threadIdx.x & 31;
  float v = enc[wid * D_ + lane];
  float ss = v * v;
#pragma unroll
  for (int o = 16; o > 0; o >>= 1) ss += __shfl_xor(ss, o, 32);
  float qn = v / fmaxf(sqrtf(ss), 1e-12f);
  int n = wid / T_;
  if (n < B_) {
    float acc = 0.0f;
#pragma unroll
    for (int dd = 0; dd < 32; ++dd) {
      float qd = __shfl(qn, dd, 32);
      float ws = 0.5f * (W[dd * 32 + lane] + W[lane * 32 + dd]);
      acc += qd * ws;
    }
    qW[wid * D_ + lane] = acc;
  } else {
    kkn[(wid - B_ * T_) * D_ + lane] = qn;
  }
}

// ------------------------------------- cost matrix GEMM via f32 WMMA (K=32)
// C[t,u,p] (p = b*R_+r) = -scale * qW[b,t,:] . kkn[r,u,:]
__global__ void k_cost_wmma(const float* __restrict__ Aq,   // [488, 32]
                            const float* __restrict__ Bk,   // [7808, 32]
                            const float* __restrict__ scale1,
                            float* __restrict__ Cmat) {     // [T_,T_,P_]
  const int MR = B_ * T_;           // 488
  const int NR = R_ * T_;           // 7808
  const int MT = (MR + 15) / 16;    // 31
  const int NTL = NR / 16;          // 488
  int wid = (blockIdx.x * blockDim.x + threadIdx.x) >> 5;
  if (wid >= MT * NTL) return;
  int tm = wid / NTL, tn = wid % NTL;
  int lane = threadIdx.x & 31;
  int hlf = lane >> 4, l16 = lane & 15;
  int m  = tm * 16 + l16;
  if (m > MR - 1) m = MR - 1;       // clamp (discarded at store)
  int nn = tn * 16 + l16;
  float sc = -scale1[0];
  v8f acc = {};
  v2f aReg[8], bReg[8];
#pragma unroll
  for (int k4 = 0; k4 < 8; ++k4) {
    int ka = k4 * 4 + hlf * 2;
    aReg[k4] = *(const v2f*)(Aq + (size_t)m * 32 + ka);
    bReg[k4] = *(const v2f*)(Bk + (size_t)nn * 32 + ka);
  }
#pragma unroll
  for (int k4 = 0; k4 < 8; ++k4)
    acc = __builtin_amdgcn_wmma_f32_16x16x4_f32(false, aReg[k4], false, bReg[k4],
                                                (short)0, acc, false, false);
  int ncol = tn * 16 + l16;
  int r = ncol / T_, u = ncol % T_;
#pragma unroll
  for (int v = 0; v < 8; ++v) {
    int mrow = tm * 16 + hlf * 8 + v;
    if (mrow < MR) {
      int b = mrow / T_, t = mrow % T_;
      Cmat[(t * T_ + u) * P_ + b * R_ + r] = sc * acc[v];
    }
  }
}

// --------------------------------------------- soft-DTW DP with _nn2 per cell
// One lane owns one (b,r) problem; prev_row lives in LDS (conflict-free).
__global__ void k_dp(const float* __restrict__ Cmat,
                     const float* __restrict__ lnw, const float* __restrict__ lnb,
                     const float* __restrict__ w1,  const float* __restrict__ b1,
                     const float* __restrict__ w2,  const float* __restrict__ b2,
                     const float* __restrict__ log_tau,
                     float* __restrict__ scores) {
  __shared__ float prev[64 * (T_ + 1)];
  int tid = threadIdx.x;                   // 0..63
  int p = blockIdx.x * 64 + tid;           // 0..1023
  // hoist nn2 params into registers (fully unrolled -> stays in VGPRs)
  float lw0 = lnw[0], lw1 = lnw[1], lw2 = lnw[2];
  float lb0 = lnb[0], lb1 = lnb[1], lb2 = lnb[2];
  float W1[16][3], Bv1[16], W2[3][16], Bv2[3];
#pragma unroll
  for (int j = 0; j < 16; ++j) {
    W1[j][0] = w1[j * 3 + 0]; W1[j][1] = w1[j * 3 + 1]; W1[j][2] = w1[j * 3 + 2];
    Bv1[j] = b1[j];
  }
#pragma unroll
  for (int i = 0; i < 3; ++i) {
    Bv2[i] = b2[i];
#pragma unroll
    for (int j = 0; j < 16; ++j) W2[i][j] = w2[i * 16 + j];
  }
  // tau = min(max(exp(log_tau), 32), 2)  (collapses to 2.0)
  float tau = fminf(fmaxf(expf(log_tau[0]), 32.0f), 2.0f);
  float rtau = 1.0f / tau;

  prev[tid] = 0.0f;
  for (int j = 1; j <= T_; ++j) prev[j * 64 + tid] = PADV;

  for (int i = 0; i < T_; ++i) {
    const float* crow = Cmat + (size_t)i * T_ * P_ + p;
    __builtin_prefetch(crow + T_ * P_, 0, 1);   // global_prefetch next row
    float diag = prev[tid];
    prev[tid] = PADV;
    float left = PADV;
    for (int j = 0; j < T_; ++j) {
      float up = prev[(j + 1) * 64 + tid];
      float c = crow[j * P_];
      float n0 = up, n1 = left, n2v = diag;
      // best = -tau * logsumexp(-n/tau) = mn - tau*log(sum exp((mn-n)/tau))
      float mn = fminf(n0, fminf(n1, n2v));
      float s = expf((mn - n0) * rtau) + expf((mn - n1) * rtau)
              + expf((mn - n2v) * rtau);
      float best = mn - tau * logf(s);
      // LN(-|n - best|)
      float a0 = -fabsf(n0 - best), a1 = -fabsf(n1 - best), a2 = -fabsf(n2v - best);
      float mu = (a0 + a1 + a2) * (1.0f / 3.0f);
      float d0 = a0 - mu, d1 = a1 - mu, d2 = a2 - mu;
      float inv = rsqrtf((d0 * d0 + d1 * d1 + d2 * d2) * (1.0f / 3.0f) + 1e-5f);
      float z0 = d0 * inv * lw0 + lb0;
      float z1 = d1 * inv * lw1 + lb1;
      float z2 = d2 * inv * lw2 + lb2;
      // 3 -> 16 (relu) -> 3
      float o0 = Bv2[0], o1 = Bv2[1], o2 = Bv2[2];
#pragma unroll
      for (int jj = 0; jj < 16; ++jj) {
        float h = fmaxf(z0 * W1[jj][0] + z1 * W1[jj][1] + z2 * W1[jj][2] + Bv1[jj], 0.0f);
        o0 += h * W2[0][jj]; o1 += h * W2[1][jj]; o2 += h * W2[2][jj];
      }
      // softmax re-weighting
      float mo = fmaxf(o0, fmaxf(o1, o2));
      float e0 = expf(o0 - mo), e1 = expf(o1 - mo), e2 = expf(o2 - mo);
      float val = c + (n0 * e0 + n1 * e1 + n2v * e2) / (e0 + e1 + e2);
      prev[(j + 1) * 64 + tid] = val;
      diag = up;
      left = val;
    }
  }
  scores[p] = prev[T_ * 64 + tid];
}

// -------------------------------------------------- soft top-k + logit head
__global__ void k_head(const float* __restrict__ scores,
                       const float* __restrict__ phi_w1, const float* __restrict__ phi_b1,
                       const float* __restrict__ phi_w2, const float* __restrict__ phi_b2,
                       const float* __restrict__ rho_w1, const float* __restrict__ rho_b1,
                       const float* __restrict__ rho_w2, const float* __restrict__ rho_b2,
                       const float* __restrict__ res1,   const float* __restrict__ res2,
                       const int* __restrict__ knn_k,
                       float* __restrict__ out) {
  int b = threadIdx.x;
  if (b >= B_) return;
  int kn = knn_k[0]; if (kn < 1) kn = 1; if (kn > 16) kn = 16;
  const float* s = scores + b * R_;

  float yk[16];
  bool used[R_];
  for (int r = 0; r < R_; ++r) used[r] = false;
  for (int k = kn - 1; k >= 0; --k) {         // yk ascending, yk[kn-1]=largest
    float best = -1e30f; int bi = 0;
    for (int r = 0; r < R_; ++r)
      if (!used[r] && s[r] > best) { best = s[r]; bi = r; }
    used[bi] = true; yk[k] = best;
  }
  // soft class assignment per selected score
  float sc[16];
  for (int k = 0; k < kn; ++k) {
    float mn = 1e30f;
    for (int r = 0; r < R_; ++r) mn = fminf(mn, fabsf(s[r] - yk[k]));
    float num = 0.0f, den = 0.0f;
    for (int r = 0; r < R_; ++r) {
      float e = expf(-(fabsf(s[r] - yk[k]) - mn) * 10.0f);   // TEMP = 0.1
      num += e * (float)(r >> 4);                            // class = r // 16
      den += e;
    }
    sc[k] = num / den;
  }
  float class_sum[KCLS_], counts[KCLS_], pooled1[KCLS_][32];
  for (int c = 0; c < KCLS_; ++c) {
    class_sum[c] = 0.0f; counts[c] = 0.0f;
    for (int j = 0; j < 32; ++j) pooled1[c][j] = 0.0f;
  }
  for (int k = 0; k < kn; ++k) {
    float dmin = 1e30f;
    for (int c = 0; c < KCLS_; ++c) dmin = fminf(dmin, fabsf(sc[k] - (float)c));
    float e[KCLS_]; float den = 0.0f;
    for (int c = 0; c < KCLS_; ++c) {
      e[c] = expf(-(fabsf(sc[k] - (float)c) - dmin) * 1e5f);
      den += e[c];
    }
    float h1v[32], h2v[32];
    for (int j = 0; j < 32; ++j) h1v[j] = gelu_exact(yk[k] * phi_w1[j] + phi_b1[j]);
    for (int j = 0; j < 32; ++j) {
      float acc = phi_b2[j];
      for (int i2 = 0; i2 < 32; ++i2) acc += h1v[i2] * phi_w2[j * 32 + i2];
      h2v[j] = gelu_exact(acc);
    }
    for (int c = 0; c < KCLS_; ++c) {
      float o = e[c] / den;
      class_sum[c] += o * yk[k];
      counts[c] += o;
      for (int j = 0; j < 32; ++j) pooled1[c][j] += o * h2v[j];
    }
  }
  for (int c = 0; c < KCLS_; ++c) {
    float t1[32];
    for (int j = 0; j < 32; ++j) {
      float acc = rho_b1[j];
      for (int i2 = 0; i2 < 32; ++i2) acc += pooled1[c][i2] * rho_w1[j * 33 + i2];
      acc += counts[c] * rho_w1[j * 33 + 32];
      t1[j] = gelu_exact(acc);
    }
    float delta = rho_b2[0];
    for (int j = 0; j < 32; ++j) delta += t1[j] * rho_w2[j];
    out[b * KCLS_ + c] = class_sum[c] * res2[0] + res1[0] * delta;
  }
}

// ---------------------------------------------------------------------------
extern "C" void kernel_launch(void* const* d_in, const int* in_sizes, int n_in,
                              void* d_out, int out_size, void* d_ws, size_t ws_size,
                              hipStream_t stream) {
  // setup_inputs() dict order, params dict flattened in insertion order:
  const float* x        = (const float*)d_in[0];
  const float* refs     = (const float*)d_in[1];
  const float* prj_w    = (const float*)d_in[2];
  const float* prj_b    = (const float*)d_in[3];
  const float* b1_dw_w  = (const float*)d_in[4];
  const float* b1_dw_b  = (const float*)d_in[5];
  const float* b1_pw_w  = (const float*)d_in[6];
  const float* b1_pw_b  = (const float*)d_in[7];
  const float* b2_dw_w  = (const float*)d_in[8];
  const float* b2_dw_b  = (const float*)d_in[9];
  const float* b2_pw_w  = (const float*)d_in[10];
  const float* b2_pw_b  = (const float*)d_in[11];
  const float* b3_dw_w  = (const float*)d_in[12];
  const float* b3_dw_b  = (const float*)d_in[13];
  const float* b3_pw_w  = (const float*)d_in[14];
  const float* b3_pw_b  = (const float*)d_in[15];
  const float* ln_w     = (const float*)d_in[16];
  const float* ln_b     = (const float*)d_in[17];
  const float* nn1_scale= (const float*)d_in[18];
  const float* nn1_W    = (const float*)d_in[19];
  const float* n2_ln_w  = (const float*)d_in[20];
  const float* n2_ln_b  = (const float*)d_in[21];
  const float* n2_w1    = (const float*)d_in[22];
  const float* n2_b1    = (const float*)d_in[23];
  const float* n2_w2    = (const float*)d_in[24];
  const float* n2_b2    = (const float*)d_in[25];
  const float* log_tau  = (const float*)d_in[26];
  const float* phi_w1   = (const float*)d_in[27];
  const float* phi_b1   = (const float*)d_in[28];
  const float* phi_w2   = (const float*)d_in[29];
  const float* phi_b2   = (const float*)d_in[30];
  const float* rho_w1   = (const float*)d_in[31];
  const float* rho_b1   = (const float*)d_in[32];
  const float* rho_w2   = (const float*)d_in[33];
  const float* rho_b2   = (const float*)d_in[34];
  const float* res1     = (const float*)d_in[35];
  const float* res2     = (const float*)d_in[36];
  const int*   knn_k    = (const int*)  d_in[37];
  float* out = (float*)d_out;

  // workspace carve-up (~28 MB of floats)
  float* ws   = (float*)d_ws;
  float* bufA = ws;                       // [NT_, 96]
  float* bufD = bufA + (size_t)NT_ * H_;  // [NT_, 96]
  float* bufB = bufD + (size_t)NT_ * H_;  // [NT_, 96]
  float* bufE = bufB + (size_t)NT_ * H_;  // [NT_, 32]
  float* enc  = bufE + (size_t)NT_ * D_;  // [NT_, 32]
  float* qWb  = enc  + (size_t)NT_ * D_;  // [488, 32]
  float* kknb = qWb  + (size_t)B_ * T_ * D_;   // [7808, 32]
  float* Cm   = kknb + (size_t)R_ * T_ * D_;   // [61, 61, 1024]
  float* scb  = Cm   + (size_t)T_ * T_ * P_;   // [1024]

  // encoder
  k_proj<<<NT_, H_, 0, stream>>>(x, refs, prj_w, prj_b, bufA);
  int dwBlocks = (NT_ * H_ + 255) / 256;
  int pw96Blocks = ((519 * 6) + 3) / 4;   // 4 waves/block of 128 threads
  int pw32Blocks = ((519 * 2) + 3) / 4;

  k_dwconv<<<dwBlocks, 256, 0, stream>>>(bufA, b1_dw_w, b1_dw_b, bufD, 1);
  k_pw_wmma<96, true><<<pw96Blocks, 128, 0, stream>>>(bufD, b1_pw_w, b1_pw_b, bufA, bufB);
  k_dwconv<<<dwBlocks, 256, 0, stream>>>(bufB, b2_dw_w, b2_dw_b, bufD, 2);
  k_pw_wmma<96, true><<<pw96Blocks, 128, 0, stream>>>(bufD, b2_pw_w, b2_pw_b, bufB, bufA);
  k_dwconv<<<dwBlocks, 256, 0, stream>>>(bufA, b3_dw_w, b3_dw_b, bufD, 4);
  k_pw_wmma<32, false><<<pw32Blocks, 128, 0, stream>>>(bufD, b3_pw_w, b3_pw_b, nullptr, bufE);

  int rowWaveBlocks = (NT_ + 3) / 4;      // one wave per row, 4 waves/block
  k_ln<<<rowWaveBlocks, 128, 0, stream>>>(bufE, ln_w, ln_b, enc);
  k_normqw<<<rowWaveBlocks, 128, 0, stream>>>(enc, nn1_W, qWb, kknb);

  // cost GEMM: 31 x 488 wave-tiles
  int costBlocks = ((31 * 488) + 3) / 4;
  k_cost_wmma<<<costBlocks, 128, 0, stream>>>(qWb, kknb, nn1_scale, Cm);

  // DP: 1024 problems, 64 threads/block
  k_dp<<<P_ / 64, 64, 0, stream>>>(Cm, n2_ln_w, n2_ln_b, n2_w1, n2_b1,
                                   n2_w2, n2_b2, log_tau, scb);

  // head
  k_head<<<1, 32, 0, stream>>>(scb, phi_w1, phi_b1, phi_w2, phi_b2,
                               rho_w1, rho_b1, rho_w2, rho_b2,
                               res1, res2, knn_k, out);
  (void)in_sizes; (void)n_in; (void)out_size; (void)ws_size;
}